// GCN_NET_22308060135605
// MI455X (gfx1250) — compile-verified
//
#include <hip/hip_runtime.h>
#include <hip/hip_bf16.h>

typedef float v2f __attribute__((ext_vector_type(2)));
typedef float v8f __attribute__((ext_vector_type(8)));

#define F1 128
#define F2 64

// ---------------------------------------------------------------- degree ----
__global__ void init_deg(float* __restrict__ deg, int n) {
    int i = blockIdx.x * blockDim.x + threadIdx.x;
    if (i < n) deg[i] = 1.0f;  // self-loop contributes 1 to every node
}

__global__ void count_deg(const long long* __restrict__ ei, float* __restrict__ deg, int E) {
    int e = blockIdx.x * blockDim.x + threadIdx.x;
    if (e < E) {
        long long c = ei[(size_t)E + e];  // col
        atomicAdd(&deg[c], 1.0f);
    }
}

__global__ void rsqrt_deg(float* __restrict__ deg, int n) {
    int i = blockIdx.x * blockDim.x + threadIdx.x;
    if (i < n) {
        float d = deg[i];
        deg[i] = (d > 0.0f) ? rsqrtf(d) : 0.0f;  // now holds dis[i]
    }
}

// ------------------------------------------------------------- WMMA GEMM ----
// C[M x N] = A[M x K] * B[K x N], fp32, one wave32 per 16x16 output tile.
// Lane layout per CDNA5 ISA 7.12.2:
//   A 16x4 : lane m=lane&15 is the row; VGPR j holds K = 2*(lane>>4) + j
//   B 4x16 : lane n=lane&15 is the col; VGPR j holds K = 2*(lane>>4) + j
//   C 16x16: VGPR v holds row M = v + 8*(lane>>4), col N = lane&15
__launch_bounds__(32)
__global__ void wmma_gemm_f32(const float* __restrict__ A,
                              const float* __restrict__ B,
                              float* __restrict__ C,
                              int K, int N) {
    const int lane = threadIdx.x & 31;
    const int half = lane >> 4;   // 0 or 1
    const int mr   = lane & 15;
    const int m0   = blockIdx.x << 4;
    const int n0   = blockIdx.y << 4;

    const float* __restrict__ arow = A + (size_t)(m0 + mr) * K;
    const float* __restrict__ bcol = B + n0 + mr;

    v8f acc = {};
    for (int k = 0; k < K; k += 4) {
        const int ka = k + (half << 1);
        v2f a, b;
        a.x = arow[ka];
        a.y = arow[ka + 1];
        b.x = bcol[(size_t)ka * N];
        b.y = bcol[(size_t)(ka + 1) * N];
        // 8 args: (neg_a, A, neg_b, B, c_mod, C, reuse_a, reuse_b)
        acc = __builtin_amdgcn_wmma_f32_16x16x4_f32(
            false, a, false, b, (short)0, acc, false, false);
    }
#pragma unroll
    for (int v = 0; v < 8; ++v) {
        C[(size_t)(m0 + (half << 3) + v) * N + (n0 + mr)] = acc[v];
    }
}

// ----------------------------------------------------- self-loop + bias -----
// agg[i,f] = dis[i]^2 * h[i,f] + bias[f]   (fully overwrites destination)
template <int F>
__global__ void self_loop_init(const float* __restrict__ dis,
                               const float* __restrict__ h,
                               const float* __restrict__ bias,
                               float* __restrict__ agg, int total) {
    int idx = blockIdx.x * blockDim.x + threadIdx.x;
    if (idx < total) {
        int i = idx / F;
        int f = idx - i * F;
        float d = dis[i];
        agg[idx] = d * d * h[idx] + bias[f];
    }
}

// ------------------------------------------------------------ edge scatter --
// One wave per edge; lane covers F/32 consecutive features.
// agg[col, :] += dis[row]*dis[col] * h[row, :]
template <int F>
__launch_bounds__(256)
__global__ void scatter_edges(const long long* __restrict__ ei,
                              const float* __restrict__ dis,
                              const float* __restrict__ h,
                              float* __restrict__ agg, int E) {
    const int lane = threadIdx.x & 31;
    const int wid  = (blockIdx.x * blockDim.x + threadIdx.x) >> 5;
    if (wid >= E) return;  // uniform per wave

    const long long r = ei[wid];
    const long long c = ei[(size_t)E + wid];
    const float nrm = dis[r] * dis[c];

    constexpr int VPT = F / 32;
    const float* __restrict__ src = h   + (size_t)r * F + lane * VPT;
    float*       __restrict__ dst = agg + (size_t)c * F + lane * VPT;
#pragma unroll
    for (int v = 0; v < VPT; ++v) {
        atomicAdd(&dst[v], nrm * src[v]);
    }
}

// ---------------------------------------------------------------- ReLU ------
__global__ void relu_inplace(float* __restrict__ x, int n) {
    int i = blockIdx.x * blockDim.x + threadIdx.x;
    if (i < n) x[i] = fmaxf(x[i], 0.0f);
}

// ---------------------------------------------------------------- launch ----
extern "C" void kernel_launch(void* const* d_in, const int* in_sizes, int n_in,
                              void* d_out, int out_size, void* d_ws, size_t ws_size,
                              hipStream_t stream) {
    const float*     x  = (const float*)d_in[0];
    const long long* ei = (const long long*)d_in[1];   // int64 edge_index [2, E]
    const float*     W1 = (const float*)d_in[2];
    const float*     b1 = (const float*)d_in[3];
    const float*     W2 = (const float*)d_in[4];
    const float*     b2 = (const float*)d_in[5];
    float* out = (float*)d_out;

    const int N = in_sizes[0] / F1;   // 50000
    const int E = in_sizes[1] / 2;    // 800000

    // Workspace layout (floats): dis[N] | bufH[N*F1] | bufG[N*F1]
    float* dis  = (float*)d_ws;
    float* bufH = dis + N;                 // gemm1 output h; reused for h2 (N*F2)
    float* bufG = bufH + (size_t)N * F1;   // aggregation buffer / h1

    const int T = 256;

    // 1) symmetric normalization coefficients
    init_deg<<<(N + T - 1) / T, T, 0, stream>>>(dis, N);
    count_deg<<<(E + T - 1) / T, T, 0, stream>>>(ei, dis, E);
    rsqrt_deg<<<(N + T - 1) / T, T, 0, stream>>>(dis, N);

    // 2) h = x @ W1   (50000x128 * 128x128)
    dim3 g1(N / 16, F1 / 16);
    wmma_gemm_f32<<<g1, 32, 0, stream>>>(x, W1, bufH, F1, F1);

    // 3) agg = dis^2*h + b1; += edge messages; ReLU
    const int nf1 = N * F1;
    self_loop_init<F1><<<(nf1 + T - 1) / T, T, 0, stream>>>(dis, bufH, b1, bufG, nf1);
    scatter_edges<F1><<<(E * 32 + T - 1) / T, T, 0, stream>>>(ei, dis, bufH, bufG, E);
    relu_inplace<<<(nf1 + T - 1) / T, T, 0, stream>>>(bufG, nf1);

    // 4) h2 = h1 @ W2   (50000x128 * 128x64), reuse bufH
    dim3 g2(N / 16, F2 / 16);
    wmma_gemm_f32<<<g2, 32, 0, stream>>>(bufG, W2, bufH, F1, F2);

    // 5) out = dis^2*h2 + b2; += edge messages
    const int nf2 = N * F2;
    self_loop_init<F2><<<(nf2 + T - 1) / T, T, 0, stream>>>(dis, bufH, b2, out, nf2);
    scatter_edges<F2><<<(E * 32 + T - 1) / T, T, 0, stream>>>(ei, dis, bufH, out, E);
}